// SortedExpertDispatch_17935783428798
// MI455X (gfx1250) — compile-verified
//
#include <hip/hip_runtime.h>

#define NUM_EXPERTS 8
#define TOP_K 2
#define NUM_TOKENS 65536
#define HIDDEN 256
// per-expert segments padded to a multiple of 16 rows
#define PERM_CAP (NUM_TOKENS + NUM_EXPERTS * 16)   // 65664
#define TILES (PERM_CAP / 16)                      // 4104

typedef __attribute__((ext_vector_type(2))) float v2f;
typedef __attribute__((ext_vector_type(8))) float v8f;

// ---------------- counting sort by primary expert ----------------

__global__ void moe_init(int* wsI, int* perm) {
    int i = blockIdx.x * blockDim.x + threadIdx.x;
    if (i < 16) wsI[i] = 0;              // counts[0..7], cursors[8..15]
    if (i < PERM_CAP) perm[i] = -1;      // padding rows marked invalid
}

__global__ void moe_hist(const int* __restrict__ eidx, int* counts) {
    int i = blockIdx.x * blockDim.x + threadIdx.x;
    if (i < NUM_TOKENS) {
        int e = eidx[i * TOP_K];         // primary expert
        atomicAdd(&counts[e], 1);
    }
}

__global__ void moe_scan(const int* __restrict__ counts, int* paddedOff) {
    if (threadIdx.x == 0 && blockIdx.x == 0) {
        int off = 0;
        for (int e = 0; e < NUM_EXPERTS; ++e) {
            paddedOff[e] = off;
            off += (counts[e] + 15) & ~15;
        }
        paddedOff[NUM_EXPERTS] = off;
    }
}

__global__ void moe_scatter(const int* __restrict__ eidx, int* cursors,
                            const int* __restrict__ paddedOff, int* perm) {
    int i = blockIdx.x * blockDim.x + threadIdx.x;
    if (i < NUM_TOKENS) {
        int e = eidx[i * TOP_K];
        int pos = atomicAdd(&cursors[e], 1);
        perm[paddedOff[e] + pos] = i;
    }
}

// ---------------- grouped GEMM via V_WMMA_F32_16X16X4_F32 ----------------
// One workgroup = 16 wave32s handles a 16-token row tile x 256 output cols.
// Each wave owns a 16x16 output tile; K-loop of 64 WMMA(16x16x4, f32).

__launch_bounds__(512)
__global__ void moe_gemm(const float* __restrict__ hs,
                         const float* __restrict__ rw,
                         const float* __restrict__ W,
                         const float* __restrict__ bias,
                         const int* __restrict__ paddedOff,
                         const int* __restrict__ perm,
                         float* __restrict__ out) {
    __shared__ float smA[16][HIDDEN + 4];   // +4 pad: conflict-free column reads
    __shared__ int   smTok[16];
    __shared__ float smRw[16];

    const int rowBase = blockIdx.x * 16;
    const int total   = paddedOff[NUM_EXPERTS];
    if (rowBase >= total) return;           // workgroup-uniform: EXEC stays full

    // which expert segment does this tile live in?
    int e = 0;
    #pragma unroll
    for (int i = 1; i < NUM_EXPERTS; ++i)
        if (rowBase >= paddedOff[i]) e = i;

    const int tid = threadIdx.x;
    if (tid < 16) {
        int tk = perm[rowBase + tid];
        smTok[tid] = tk;
        smRw[tid]  = (tk >= 0) ? rw[tk * TOP_K] : 0.0f;
    }
    // stage A tile: 16 tokens x 256 f32 (4096 elems, 8 per thread)
    for (int i = tid; i < 16 * HIDDEN; i += 512) {
        int r = i >> 8, c = i & (HIDDEN - 1);
        int tk = perm[rowBase + r];
        smA[r][c] = (tk >= 0) ? hs[(size_t)tk * HIDDEN + c] : 0.0f;
    }
    __syncthreads();

    const int wave    = tid >> 5;           // 0..15 -> column strip
    const int lane    = tid & 31;
    const int colBase = wave * 16;
    const int n       = lane & 15;          // A row (M) and B/C column (N)
    const int hi      = lane >> 4;          // selects K pair {0,1} vs {2,3}

    const float* wp = W + (size_t)e * HIDDEN * HIDDEN + colBase + n;

    v8f acc = {};
    for (int k = 0; k < HIDDEN; k += 4) {
        const int ka = k + hi * 2;
        v2f a, b;
        // A 16x4 f32 layout: lanes 0-15 -> M=lane, K=0/1; lanes 16-31 -> M=lane-16, K=2/3
        a.x = smA[n][ka];
        a.y = smA[n][ka + 1];
        // B 4x16 f32 layout (mirror): VGPR0 = K=(hi?2:0), VGPR1 = K=(hi?3:1), N = lane%16
        b.x = wp[(size_t)ka * HIDDEN];
        b.y = wp[(size_t)(ka + 1) * HIDDEN];
        acc = __builtin_amdgcn_wmma_f32_16x16x4_f32(
            /*neg_a=*/false, a, /*neg_b=*/false, b,
            /*c_mod=*/(short)0, acc, /*reuse_a=*/false, /*reuse_b=*/false);
    }

    const float bval = bias[(size_t)e * HIDDEN + colBase + n];
    // C/D layout: VGPR v -> M = v + 8*hi, N = lane%16
    #pragma unroll
    for (int v = 0; v < 8; ++v) {
        int m  = v + hi * 8;
        int tk = smTok[m];
        if (tk >= 0)
            out[(size_t)tk * HIDDEN + colBase + n] = (acc[v] + bval) * smRw[m];
    }
}

// ---------------- launcher ----------------

extern "C" void kernel_launch(void* const* d_in, const int* in_sizes, int n_in,
                              void* d_out, int out_size, void* d_ws, size_t ws_size,
                              hipStream_t stream) {
    const float* hs   = (const float*)d_in[0];   // (65536, 256) f32
    const int*   eidx = (const int*)  d_in[1];   // (65536, 2)   i32
    const float* rw   = (const float*)d_in[2];   // (65536, 2)   f32
    const float* W    = (const float*)d_in[3];   // (8, 256, 256) f32
    const float* b    = (const float*)d_in[4];   // (8, 256)     f32
    float* out = (float*)d_out;

    int* wsI       = (int*)d_ws;
    int* counts    = wsI + 0;
    int* cursors   = wsI + 8;
    int* paddedOff = wsI + 16;   // 9 entries
    int* perm      = wsI + 32;   // PERM_CAP entries

    moe_init<<<(PERM_CAP + 255) / 256, 256, 0, stream>>>(wsI, perm);
    moe_hist<<<NUM_TOKENS / 256, 256, 0, stream>>>(eidx, counts);
    moe_scan<<<1, 32, 0, stream>>>(counts, paddedOff);
    moe_scatter<<<NUM_TOKENS / 256, 256, 0, stream>>>(eidx, cursors, paddedOff, perm);
    moe_gemm<<<TILES, 512, 0, stream>>>(hs, rw, W, b, paddedOff, perm, out);
}